// MMDLoss_64433099374604
// MI455X (gfx1250) — compile-verified
//
#include <hip/hip_runtime.h>

typedef __attribute__((ext_vector_type(2))) float v2f;
typedef __attribute__((ext_vector_type(8))) float v8f;

#define NX   4096
#define NTOT 8192
#define DDIM 64
// N^2 - N for N = 8192
#define DENOM_BW 67100672.0

__device__ __forceinline__ const float* zrow(const float* X, const float* Y, int r) {
    return (r < NX) ? (X + (size_t)r * DDIM) : (Y + (size_t)(r - NX) * DDIM);
}

// ---------------------------------------------------------------------------
// Kernel 0: per-row squared norms sq[i] = sum_k Z[i][k]^2
// ---------------------------------------------------------------------------
__global__ __launch_bounds__(256) void k_sqnorm(const float* __restrict__ X,
                                                const float* __restrict__ Y,
                                                float* __restrict__ sq) {
    int i = blockIdx.x * blockDim.x + threadIdx.x;
    if (i < NTOT) {
        const float* p = zrow(X, Y, i);
        float s = 0.f;
#pragma unroll
        for (int k = 0; k < DDIM; k += 4) {
            float4 v = *(const float4*)(p + k);
            s += v.x * v.x + v.y * v.y + v.z * v.z + v.w * v.w;
        }
        sq[i] = s;
    }
}

// ---------------------------------------------------------------------------
// Tile kernel: each wave computes a 32x32 macro-tile of the Gram matrix with
// V_WMMA_F32_16X16X4_F32 (2x2 sub-tiles, 16 K-steps of K=4).
// PASS 0: accumulate sum(max(L2,0))          -> bandwidth numerator
// PASS 1: accumulate sum_k exp(-L2/(bw*m_k)) -> kernel-block sums
// One double partial per workgroup (deterministic fixed-order reduction).
// ---------------------------------------------------------------------------
template <int PASS>
__global__ __launch_bounds__(256) void k_tiles(const float* __restrict__ X,
                                               const float* __restrict__ Y,
                                               const float* __restrict__ sq,
                                               const double* __restrict__ acc,
                                               double* __restrict__ parts) {
    const int lane  = threadIdx.x & 31;
    const int wave  = threadIdx.x >> 5;
    const int tilex = blockIdx.x * 8 + wave;   // 0..255 (8 waves/WG along cols)
    const int tiley = blockIdx.y;              // 0..255
    const int rm = tiley * 32;                 // macro-tile row base
    const int cn = tilex * 32;                 // macro-tile col base
    const int m     = lane & 15;
    const int khalf = lane >> 4;

    // A fragment rows (rm.., rm+16..) and B fragment rows (cn.., cn+16..).
    const float* ar0 = zrow(X, Y, rm + m);
    const float* ar1 = zrow(X, Y, rm + 16 + m);
    const float* br0 = zrow(X, Y, cn + m);
    const float* br1 = zrow(X, Y, cn + 16 + m);

    v8f c00 = {}, c01 = {}, c10 = {}, c11 = {};
#pragma unroll
    for (int s = 0; s < 16; ++s) {
        const int k0 = 4 * s + 2 * khalf;      // two contiguous K values per lane
        v2f a0 = *(const v2f*)(ar0 + k0);
        v2f a1 = *(const v2f*)(ar1 + k0);
        v2f b0 = *(const v2f*)(br0 + k0);
        v2f b1 = *(const v2f*)(br1 + k0);
        c00 = __builtin_amdgcn_wmma_f32_16x16x4_f32(false, a0, false, b0, (short)0, c00, false, false);
        c01 = __builtin_amdgcn_wmma_f32_16x16x4_f32(false, a0, false, b1, (short)0, c01, false, false);
        c10 = __builtin_amdgcn_wmma_f32_16x16x4_f32(false, a1, false, b0, (short)0, c10, false, false);
        c11 = __builtin_amdgcn_wmma_f32_16x16x4_f32(false, a1, false, b1, (short)0, c11, false, false);
    }

    // Row/col squared norms for this lane's D-matrix elements.
    float sqr0[8], sqr1[8];
#pragma unroll
    for (int v = 0; v < 8; ++v) {
        sqr0[v] = sq[rm + 8 * khalf + v];       // rows of c00/c01
        sqr1[v] = sq[rm + 16 + 8 * khalf + v];  // rows of c10/c11
    }
    const float sqc0 = sq[cn + m];              // cols of c00/c10
    const float sqc1 = sq[cn + 16 + m];         // cols of c01/c11

    float cexp = 0.f;
    if (PASS == 1) cexp = (float)acc[5];        // -0.25 / bandwidth (precomputed)

    float partial = 0.f;
#pragma unroll
    for (int v = 0; v < 8; ++v) {
        float l2a = fmaxf(sqr0[v] + sqc0 - 2.0f * c00[v], 0.0f);
        float l2b = fmaxf(sqr0[v] + sqc1 - 2.0f * c01[v], 0.0f);
        float l2c = fmaxf(sqr1[v] + sqc0 - 2.0f * c10[v], 0.0f);
        float l2d = fmaxf(sqr1[v] + sqc1 - 2.0f * c11[v], 0.0f);
        if (PASS == 0) {
            partial += (l2a + l2b) + (l2c + l2d);
        } else {
            // sum_k exp(-L2/(bw*2^(k-2))) = t + t^2 + t^4 + t^8 + t^16,
            // t = exp(-L2/(4*bw)) : one v_exp_f32 instead of five.
            float ta = __expf(l2a * cexp), tb = __expf(l2b * cexp);
            float tc = __expf(l2c * cexp), td = __expf(l2d * cexp);
            float ta2 = ta * ta, ta4 = ta2 * ta2, ta8 = ta4 * ta4, ta16 = ta8 * ta8;
            float tb2 = tb * tb, tb4 = tb2 * tb2, tb8 = tb4 * tb4, tb16 = tb8 * tb8;
            float tc2 = tc * tc, tc4 = tc2 * tc2, tc8 = tc4 * tc4, tc16 = tc8 * tc8;
            float td2 = td * td, td4 = td2 * td2, td8 = td4 * td4, td16 = td8 * td8;
            partial += (ta + ta2 + ta4 + ta8 + ta16) + (tb + tb2 + tb4 + tb8 + tb16);
            partial += (tc + tc2 + tc4 + tc8 + tc16) + (td + td2 + td4 + td8 + td16);
        }
    }

    // Fixed-pattern wave reduction (deterministic).
#pragma unroll
    for (int off = 16; off > 0; off >>= 1)
        partial += __shfl_xor(partial, off, 32);

    __shared__ float red[8];
    if (lane == 0) red[wave] = partial;
    __syncthreads();
    if (threadIdx.x == 0) {
        double sblk = 0.0;
        for (int w = 0; w < 8; ++w) sblk += (double)red[w];  // fixed order
        parts[blockIdx.y * 32 + blockIdx.x] = sblk;
    }
}

// ---------------------------------------------------------------------------
// Deterministic single-block reduction of the 8192 per-WG partials.
// mode 0: acc[0] = total; acc[5] = -0.25/bandwidth
// mode 1: acc[1..4] = block sums (XX, XY, YX, YY)
// ---------------------------------------------------------------------------
__global__ __launch_bounds__(256) void k_reduce(const double* __restrict__ parts,
                                                double* __restrict__ acc, int mode) {
    __shared__ double lds[256][4];
    const int tid = threadIdx.x;
    double s[4] = {0.0, 0.0, 0.0, 0.0};
    for (int p = tid; p < 8192; p += 256) {      // fixed per-thread schedule
        int bx = p & 31;                         // column workgroup index
        int by = p >> 5;                         // row tile index
        int bin = 2 * (by >= 128) + (bx >= 16);  // (row in Y, col in Y)
        s[bin] += parts[p];
    }
#pragma unroll
    for (int j = 0; j < 4; ++j) lds[tid][j] = s[j];
    __syncthreads();
    for (int stride = 128; stride > 0; stride >>= 1) {
        if (tid < stride) {
#pragma unroll
            for (int j = 0; j < 4; ++j) lds[tid][j] += lds[tid + stride][j];
        }
        __syncthreads();
    }
    if (tid == 0) {
        if (mode == 0) {
            double tot = ((lds[0][0] + lds[0][1]) + (lds[0][2] + lds[0][3]));
            acc[0] = tot;
            double bw = tot / DENOM_BW;          // median-heuristic bandwidth
            acc[5] = -0.25 / bw;                 // exp scale for t = exp(-L2/(4bw))
        } else {
            acc[1] = lds[0][0];
            acc[2] = lds[0][1];
            acc[3] = lds[0][2];
            acc[4] = lds[0][3];
        }
    }
}

// ---------------------------------------------------------------------------
// Final scalar: mean(K_XX) - 2*mean(K_XY) + mean(K_YY)
// ---------------------------------------------------------------------------
__global__ void k_final(const double* __restrict__ acc, float* __restrict__ out) {
    if (threadIdx.x == 0 && blockIdx.x == 0) {
        const double denom = (double)NX * (double)NX;
        double XXm = acc[1] / denom;
        double XYm = acc[2] / denom;   // K[:nX, nX:] block, as in reference
        double YYm = acc[4] / denom;
        out[0] = (float)(XXm - 2.0 * XYm + YYm);
    }
}

extern "C" void kernel_launch(void* const* d_in, const int* in_sizes, int n_in,
                              void* d_out, int out_size, void* d_ws, size_t ws_size,
                              hipStream_t stream) {
    (void)in_sizes; (void)n_in; (void)out_size; (void)ws_size;
    const float* X = (const float*)d_in[0];
    const float* Y = (const float*)d_in[1];
    float* out = (float*)d_out;

    // Workspace layout (all freshly overwritten each call; no init needed):
    //   [0,   48)  : double acc[6]  (sum, S00..S11, exp-scale)
    //   [64,  32832): float  sq[8192]
    //   [32832, 98368): double parts[8192]
    char* ws = (char*)d_ws;
    double* acc   = (double*)(ws);
    float*  sq    = (float*)(ws + 64);
    double* parts = (double*)(ws + 64 + NTOT * sizeof(float));

    dim3 tgrid(32, 256);   // 8 col-tiles per WG (8 waves) x 256 row-tiles

    k_sqnorm<<<NTOT / 256, 256, 0, stream>>>(X, Y, sq);
    k_tiles<0><<<tgrid, 256, 0, stream>>>(X, Y, sq, acc, parts);
    k_reduce<<<1, 256, 0, stream>>>(parts, acc, 0);
    k_tiles<1><<<tgrid, 256, 0, stream>>>(X, Y, sq, acc, parts);
    k_reduce<<<1, 256, 0, stream>>>(parts, acc, 1);
    k_final<<<1, 1, 0, stream>>>(acc, out);
}